// Net_31997506355959
// MI455X (gfx1250) — compile-verified
//
#include <hip/hip_runtime.h>

typedef __attribute__((ext_vector_type(2))) float v2f;
typedef __attribute__((ext_vector_type(8))) float v8f;
typedef __attribute__((__vector_size__(2 * sizeof(int)))) int v2i;

#define NI 784
#define NH 300
#define NO 10
#define NSTEP 100
#define KJ 75            // NH / 4  (K-steps of the f32 WMMA)
#define SLOT 152         // dwords per lane fragment slot (75 * 2 + 2 pad)
#define NTILES 19        // ceil(NH / 16)

#if defined(__has_builtin)
#if __has_builtin(__builtin_amdgcn_global_load_async_to_lds_b64) && \
    __has_builtin(__builtin_amdgcn_s_wait_asynccnt)
#define HAVE_ASYNC_LDS 1
typedef __attribute__((address_space(1))) v2i g_v2i;
typedef __attribute__((address_space(3))) v2i l_v2i;
#endif
#endif

// ---------------------------------------------------------------------------
// Kernel 1: cur1 = x @ w1^T + b1   via V_WMMA_F32_16X16X4_F32
// 8 waves of a block share one A tile (same mTile, 8 consecutive nTiles) so
// identical x loads collapse in WGP$/L2. Dual accumulator chains for WMMA ILP.
// ---------------------------------------------------------------------------
__global__ void __launch_bounds__(256)
snn_fc1_wmma(const float* __restrict__ x, const float* __restrict__ w1,
             const float* __restrict__ b1, float* __restrict__ cur1,
             int mTiles)
{
  const int lane = threadIdx.x & 31;
  const int wave = threadIdx.x >> 5;
  int mTile = blockIdx.x % mTiles;
  int nTile = (blockIdx.x / mTiles) * 8 + wave;
  if (nTile >= NTILES) return;

  const int col = lane & 15;
  const int hi  = lane >> 4;
  int n    = nTile * 16 + col;
  int nC   = (n < NH) ? n : (NH - 1);     // clamp OOB loads, stores are guarded
  int mRow = mTile * 16 + col;

  const float* aPtr = x  + (size_t)mRow * NI + 2 * hi;
  const float* bPtr = w1 + (size_t)nC   * NI + 2 * hi;

  float bias = (n < NH) ? b1[n] : 0.0f;   // same col for both lane halves
  v8f c0 = {bias, bias, bias, bias, bias, bias, bias, bias};
  v8f c1 = {0.f, 0.f, 0.f, 0.f, 0.f, 0.f, 0.f, 0.f};

  #pragma unroll 2
  for (int k = 0; k < NI; k += 8) {       // 784 = 98 * 8, exact
    v2f a0 = *(const v2f*)(aPtr + k);
    v2f b0 = *(const v2f*)(bPtr + k);
    v2f a1 = *(const v2f*)(aPtr + k + 4);
    v2f b1v = *(const v2f*)(bPtr + k + 4);
    c0 = __builtin_amdgcn_wmma_f32_16x16x4_f32(false, a0, false, b0,
                                               (short)0, c0, false, false);
    c1 = __builtin_amdgcn_wmma_f32_16x16x4_f32(false, a1, false, b1v,
                                               (short)0, c1, false, false);
  }
  v8f acc = c0 + c1;

  if (n < NH) {
    int rowBase = mTile * 16 + (hi ? 8 : 0);   // C layout: vgpr r -> M=r / r+8
    #pragma unroll
    for (int r = 0; r < 8; ++r)
      cur1[(size_t)(rowBase + r) * NH + n] = acc[r];
  }
}

// ---------------------------------------------------------------------------
// Kernel 2: 100-step leaky-integrate-fire recurrence.
// 2 waves / 32 batch rows per workgroup; ~512 waves total (= parallelism cap
// of 8192/16 M-tiles), so ~1 wave/SIMD: optimize for per-wave ILP.
//   - m1 state AND cur1 fragments live in registers (A-fragment layout)
//   - w2 fragments staged once into LDS via async global->LDS copy
//   - two interleaved WMMA accumulator chains break the serial C dependency
// ---------------------------------------------------------------------------
__global__ void __launch_bounds__(64)
snn_steps_wmma(const float* __restrict__ cur1, const float* __restrict__ w2,
               const float* __restrict__ b2, const float* __restrict__ beta1,
               const float* __restrict__ beta2,
               float* __restrict__ spkOut, float* __restrict__ memOut, int B)
{
  __shared__ __align__(16) float w2F[32 * SLOT];   // per-lane w2 fragments

  const int tid  = threadIdx.x;
  const int lane = tid & 31;
  const int wave = tid >> 5;
  const int col  = lane & 15;
  const int hi   = lane >> 4;
  const int rowBase = blockIdx.x * 32 + wave * 16;

  // Stage w2 fragments into LDS (shared by both waves): B[k][n] = w2[n][k],
  // zero-pad n >= 10. Async global->LDS copy where available.
  if (wave == 0) {
    float* dst = w2F + lane * SLOT;
    if (col < NO) {
      const float* src = w2 + (size_t)col * NH + 2 * hi;
#if defined(HAVE_ASYNC_LDS)
      #pragma unroll 5
      for (int j = 0; j < KJ; ++j)
        __builtin_amdgcn_global_load_async_to_lds_b64(
            (g_v2i*)(src + 4 * j),
            (l_v2i*)(dst + 2 * j),
            0, 0);
#else
      #pragma unroll 5
      for (int j = 0; j < KJ; ++j)
        *(v2f*)(dst + 2 * j) = *(const v2f*)(src + 4 * j);
#endif
    } else {
      v2f z = {0.f, 0.f};
      #pragma unroll 5
      for (int j = 0; j < KJ; ++j)
        *(v2f*)(dst + 2 * j) = z;
    }
  }

  // cur1 fragments -> registers: lane owns row (rowBase+col), k = 4j+2hi+{0,1}
  v2f curf[KJ];
  {
    const float* src = cur1 + (size_t)(rowBase + col) * NH + 2 * hi;
    #pragma unroll
    for (int j = 0; j < KJ; ++j)
      curf[j] = *(const v2f*)(src + 4 * j);
  }

  float be1 = fminf(fmaxf(beta1[0], 0.f), 1.f);
  float be2v = 0.f, b2v = 0.f;
  if (col < NO) {
    be2v = fminf(fmaxf(beta2[col], 0.f), 1.f);
    b2v  = b2[col];
  }

#if defined(HAVE_ASYNC_LDS)
  __builtin_amdgcn_s_wait_asynccnt(0);   // uniform wait; 0 outstanding on wave1
#endif
  __syncthreads();

  v2f m1f[KJ];                                  // membrane state, registers
  #pragma unroll
  for (int j = 0; j < KJ; ++j) m1f[j] = (v2f){0.f, 0.f};
  v8f m2 = {0.f, 0.f, 0.f, 0.f, 0.f, 0.f, 0.f, 0.f};

  const int orow0 = rowBase + (hi ? 8 : 0);
  const bool doStore = (col < NO);
  const size_t stepStride = (size_t)B * NO;
  float* spkP = spkOut + (size_t)orow0 * NO + col;
  float* memP = memOut + (size_t)orow0 * NO + col;
  const float* wf = w2F + lane * SLOT;

  for (int step = 0; step < NSTEP; ++step) {
    v8f c0 = {0.f, 0.f, 0.f, 0.f, 0.f, 0.f, 0.f, 0.f};
    v8f c1 = {0.f, 0.f, 0.f, 0.f, 0.f, 0.f, 0.f, 0.f};
    #pragma unroll
    for (int j = 0; j < KJ; ++j) {
      v2f wv = *(const v2f*)(wf + 2 * j);
      v2f cu = curf[j];
      v2f m  = m1f[j];
      float rx = (m.x > 1.0f) ? 1.0f : 0.0f;    // reset from previous membrane
      float ry = (m.y > 1.0f) ? 1.0f : 0.0f;
      m.x = be1 * m.x + cu.x - rx;
      m.y = be1 * m.y + cu.y - ry;
      m1f[j] = m;
      v2f s;                                    // spike = (m - 1 > 0)
      s.x = (m.x > 1.0f) ? 1.0f : 0.0f;
      s.y = (m.y > 1.0f) ? 1.0f : 0.0f;
      if (j & 1)
        c1 = __builtin_amdgcn_wmma_f32_16x16x4_f32(false, s, false, wv,
                                                   (short)0, c1, false, false);
      else
        c0 = __builtin_amdgcn_wmma_f32_16x16x4_f32(false, s, false, wv,
                                                   (short)0, c0, false, false);
    }
    v8f c = c0 + c1;
    const size_t so = (size_t)step * stepStride;
    #pragma unroll
    for (int r = 0; r < 8; ++r) {
      float m2v = m2[r];
      float rst = (m2v > 1.0f) ? 1.0f : 0.0f;
      m2v = be2v * m2v + (c[r] + b2v) - rst;
      m2[r] = m2v;
      float s2 = (m2v > 1.0f) ? 1.0f : 0.0f;
      if (doStore) {
        spkP[so + r * NO] = s2;
        memP[so + r * NO] = m2v;
      }
    }
  }
}

// ---------------------------------------------------------------------------
extern "C" void kernel_launch(void* const* d_in, const int* in_sizes, int n_in,
                              void* d_out, int out_size, void* d_ws, size_t ws_size,
                              hipStream_t stream) {
  (void)n_in; (void)out_size; (void)ws_size;
  const float* x     = (const float*)d_in[0];
  const float* w1    = (const float*)d_in[1];
  const float* b1    = (const float*)d_in[2];
  const float* beta1 = (const float*)d_in[3];
  const float* w2    = (const float*)d_in[4];
  const float* b2    = (const float*)d_in[5];
  const float* beta2 = (const float*)d_in[6];

  const int B = in_sizes[0] / NI;               // 8192
  float* cur1 = (float*)d_ws;                   // [B, NH] f32 scratch (9.8 MB)
  float* spk  = (float*)d_out;                  // [100, B, 10]
  float* memr = spk + (size_t)NSTEP * B * NO;   // [100, B, 10]

  int mTiles  = B / 16;                         // 512
  int nBlk    = (NTILES + 7) / 8;               // 3 blocks of 8 nTiles
  int blocks1 = mTiles * nBlk;                  // 1536

  snn_fc1_wmma<<<blocks1, 256, 0, stream>>>(x, w1, b1, cur1, mTiles);
  snn_steps_wmma<<<B / 32, 64, 0, stream>>>(cur1, w2, b2, beta1, beta2,
                                            spk, memr, B);
}